// FaithfulSTAN_46420006536006
// MI455X (gfx1250) — compile-verified
//
#include <hip/hip_runtime.h>
#include <hip/hip_bf16.h>
#include <math.h>

// ---------------------------------------------------------------------------
// FaithfulSTAN on gfx1250 (CDNA5): fp32 WMMA (V_WMMA_F32_16X16X4_F32) for all
// GEMMs, GLOBAL_LOAD_ASYNC_TO_LDS_B128 (+ s_wait_asynccnt) tile staging,
// TRANS-unit haversines overlapping XDL, fused matcher softmax.
// Padded layout: each batch occupies 64 rows (n in [0,64), valid n < 50).
// ---------------------------------------------------------------------------

typedef __attribute__((ext_vector_type(2))) float v2f;
typedef __attribute__((ext_vector_type(8))) float v8f;

#define BB      1024
#define NN      50
#define NPAD    64
#define DD      128
#define RR      1000
#define NPOIS   50000
#define TMAXC   10080.0f
#define DMAXC   200.0f
#define RSQRT_D 0.08838834764831845f   // 1/sqrt(128)
#define NEG_BIG (-1.0e30f)

__device__ __forceinline__ float hav_km(float lat1, float lon1, float lat2, float lon2) {
    const float DEG2RAD = 0.017453292519943295f;
    float dlat = (lat2 - lat1) * DEG2RAD;
    float dlon = (lon2 - lon1) * DEG2RAD;
    float sa = __sinf(0.5f * dlat);
    float sb = __sinf(0.5f * dlon);
    float a = sa * sa + __cosf(lat1 * DEG2RAD) * __cosf(lat2 * DEG2RAD) * sb * sb;
    a = fminf(fmaxf(a, 0.0f), 1.0f);
    return 6371.0f * 2.0f * asinf(sqrtf(a));
}

// linear interp into a 64-entry table, matching _interp_scalar
__device__ __forceinline__ float interp64(const float* tab, float x) {
    x = fminf(fmaxf(x, 0.0f), 63.0f - 1e-6f);
    int   k = (int)floorf(x);
    float f = x - (float)k;
    int   k1 = (k + 1 < 63) ? (k + 1) : 63;
    return (1.0f - f) * tab[k] + f * tab[k1];
}

// D = A(16x4) * B(4x16) + C, fp32 WMMA fragment step.
// A lane layout: lanes0-15 M=0..15; v0 holds K=(0|2), v1 holds K=(1|3) per half.
// B mirrored: v0 K-row (0|2), v1 K-row (1|3), N = lane%16.
__device__ __forceinline__ v8f wmma_f32_step(v2f a, v2f b, v8f c) {
    return __builtin_amdgcn_wmma_f32_16x16x4_f32(false, a, false, b, (short)0, c, false, false);
}

// --- CDNA5 async global->LDS copy (ASYNCcnt tracked, §10/§15.18) -----------
// Flat LDS addresses carry the LDS byte offset in addr[31:0] (aperture map,
// ISA §10.2), so truncating the flat pointer yields the VDST LDS address.
__device__ __forceinline__ uint32_t lds_addr_of(const void* p) {
    return (uint32_t)(uintptr_t)p;
}
__device__ __forceinline__ void async_copy_b128(uint32_t lds_byte_addr, const void* gptr) {
    asm volatile("global_load_async_to_lds_b128 %0, %1, off"
                 :: "v"(lds_byte_addr), "v"((uint64_t)(uintptr_t)gptr)
                 : "memory");
}
__device__ __forceinline__ void wait_async0() {
    asm volatile("s_wait_asynccnt 0x0" ::: "memory");
}

// ---------------------------------------------------------------------------
// Kernel 1: x = poi_emb[poi] + time_emb[hour] (padded rows -> 0), then
// Q = x Wq^T, K = x Wk^T, V = x Wv^T via WMMA. 16 rows per block, 4 waves.
// Weight matrices are 64 KB x3 and L2-resident; per-step global_load_b64
// fragments pipeline against WMMA under s_clause.
// ---------------------------------------------------------------------------
__global__ __launch_bounds__(128) void k_embed_qkv(
    const int* __restrict__ poi_idx, const int* __restrict__ hour,
    const float* __restrict__ poi_emb, const float* __restrict__ time_emb,
    const float* __restrict__ Wq, const float* __restrict__ Wk, const float* __restrict__ Wv,
    float* __restrict__ Q, float* __restrict__ K, float* __restrict__ V)
{
    __shared__ float xs[16 * DD];
    const int tid = threadIdx.x;
    const int rowBase = blockIdx.x * 16;

    // build the 16x128 x-tile in LDS (gather + add; invalid padded rows -> 0)
    for (int i = tid; i < 16 * DD; i += 128) {
        int r = i >> 7, c = i & 127;
        int rp = rowBase + r;
        int b = rp >> 6, n = rp & (NPAD - 1);
        float v = 0.0f;
        if (n < NN) {
            int pi = poi_idx[b * NN + n];
            bool pad = pi < 0;
            int ps = pad ? NPOIS : pi;
            int hs = pad ? 0 : hour[b * NN + n];
            v = poi_emb[(size_t)ps * DD + c] + time_emb[(size_t)hs * DD + c];
        }
        xs[i] = v;
    }
    __syncthreads();

    const int wave = tid >> 5, lane = tid & 31;
    const int half = lane >> 4, l16 = lane & 15;

    // 24 output tiles = 3 matrices x 8 col-tiles; 6 per wave
    for (int ti = 0; ti < 6; ++ti) {
        int t = wave * 6 + ti;
        const float* W = (t < 8) ? Wq : (t < 16) ? Wk : Wv;
        float* Out     = (t < 8) ? Q  : (t < 16) ? K  : V;
        int colBase = (t & 7) * 16;

        const float* wrow = W + (size_t)(colBase + l16) * DD;
        __builtin_prefetch(wrow, 0, 3);   // global_prefetch_b8 (L2-resident weights)

        v8f acc = {};
        for (int s = 0; s < 32; ++s) {
            int k = 4 * s + 2 * half;
            v2f a, b;
            a.x = xs[l16 * DD + k];
            a.y = xs[l16 * DD + k + 1];
            b.x = wrow[k];          // B[k][n] = W[n][k]  (x @ W^T)
            b.y = wrow[k + 1];
            acc = wmma_f32_step(a, b, acc);
        }
#pragma unroll
        for (int j = 0; j < 8; ++j) {
            int row = rowBase + j + 8 * half;
            Out[(size_t)row * DD + colBase + l16] = acc[j];
        }
    }
}

// ---------------------------------------------------------------------------
// Kernel 2: per-batch attention. scores = QK^T/sqrt(D) + bias(dt,dd), mask,
// softmax over keys, S = attn @ V. One block (8 waves) per batch, N padded 64.
// V tile (32 KB) is async-staged to LDS at kernel start; the copy overlaps
// the scores WMMA + haversine bias + softmax, and is only waited on before
// the attn @ V phase.
// ---------------------------------------------------------------------------
__global__ __launch_bounds__(256) void k_attn(
    const int* __restrict__ poi_idx,
    const float* __restrict__ lat, const float* __restrict__ lon, const float* __restrict__ t_min,
    const float* __restrict__ Et, const float* __restrict__ Ed,
    const float* __restrict__ Q, const float* __restrict__ Km, const float* __restrict__ V,
    float* __restrict__ S)
{
    __shared__ float sc[NPAD * NPAD];                 // scores -> attn (16 KB)
    __shared__ float sV[NPAD * DD];                   // V tile (32 KB)
    __shared__ float slat[NPAD], slon[NPAD], st[NPAD], sval[NPAD];
    __shared__ float sEt[64], sEd[64];

    const int b = blockIdx.x;
    const int tid = threadIdx.x;

    // --- async prefetch of V[b] into LDS (ASYNCcnt) ------------------------
    {
        const float* vsrc = V + (size_t)b * NPAD * DD;
        for (int c = tid; c < (NPAD * DD) / 4; c += 256)       // 2048 x b128
            async_copy_b128(lds_addr_of(&sV[c * 4]), vsrc + c * 4);
    }

    if (tid < 64) {
        sEt[tid] = Et[tid];
        sEd[tid] = Ed[tid];
        float vl = 0.f, la = 0.f, lo = 0.f, tm = 0.f;
        if (tid < NN) {
            int pi = poi_idx[b * NN + tid];
            vl = (pi >= 0) ? 1.0f : 0.0f;
            la = lat[b * NN + tid];
            lo = lon[b * NN + tid];
            tm = t_min[b * NN + tid];
        }
        sval[tid] = vl; slat[tid] = la; slon[tid] = lo; st[tid] = tm;
    }
    __syncthreads();

    const int wave = tid >> 5, lane = tid & 31;
    const int half = lane >> 4, l16 = lane & 15;

    // --- scores = Q K^T * 1/sqrt(D) : 16 tiles of 16x16, 2 per wave ---------
    for (int tt = 0; tt < 2; ++tt) {
        int t = wave + tt * 8;
        int nT = t >> 2, mT = t & 3;
        const float* qrow = Q  + (size_t)(b * NPAD + nT * 16 + l16) * DD;
        const float* krow = Km + (size_t)(b * NPAD + mT * 16 + l16) * DD;
        v8f acc = {};
        for (int s = 0; s < 32; ++s) {
            int k = 4 * s + 2 * half;
            v2f a, bf;
            a.x = qrow[k];  a.y = qrow[k + 1];
            bf.x = krow[k]; bf.y = krow[k + 1];   // B[k][m] = K[m][k]
            acc = wmma_f32_step(a, bf, acc);
        }
#pragma unroll
        for (int j = 0; j < 8; ++j)
            sc[(nT * 16 + j + 8 * half) * NPAD + mT * 16 + l16] = acc[j] * RSQRT_D;
    }
    __syncthreads();

    // --- bias (interp(E_t, dt) + interp(E_d, dd)) + key mask ---------------
    for (int i = 0; i < 16; ++i) {
        int idx = tid + i * 256;
        int n = idx >> 6, m = idx & (NPAD - 1);
        float vp = sval[n] * sval[m];                          // 0 beyond N or pad
        float dt = fabsf(st[n] - st[m]) * vp;
        float dd = hav_km(slat[n], slon[n], slat[m], slon[m]) * vp;
        float tb = fminf(dt, TMAXC) / TMAXC * 63.0f;
        float db = fminf(fmaxf(dd, 0.0f), DMAXC) / DMAXC * 63.0f;
        float bias = interp64(sEt, tb) + interp64(sEd, db);
        bool validm = (m < NN) && (sval[m] > 0.0f);
        sc[idx] = validm ? (sc[idx] + bias) : NEG_BIG;
    }
    __syncthreads();

    // --- row softmax over keys (64 rows, one thread each) ------------------
    if (tid < NPAD) {
        float mx = NEG_BIG;
        for (int m = 0; m < NPAD; ++m) mx = fmaxf(mx, sc[tid * NPAD + m]);
        float sum = 0.0f;
        for (int m = 0; m < NPAD; ++m) {
            float e = __expf(sc[tid * NPAD + m] - mx);
            sc[tid * NPAD + m] = e;
            sum += e;
        }
        float inv = (sum > 0.0f && mx > -1e29f) ? 1.0f / sum : 0.0f;
        for (int m = 0; m < NPAD; ++m) sc[tid * NPAD + m] *= inv;
    }

    // V tile must be resident before phase D: each wave drains its own
    // ASYNCcnt, then the barrier publishes the LDS tile workgroup-wide.
    wait_async0();
    __syncthreads();

    // --- S = attn @ V : 32 tiles (4 nT x 8 dT), 4 per wave, K=64 -----------
    for (int tt = 0; tt < 4; ++tt) {
        int t = wave * 4 + tt;
        int nT = t >> 3, dT = t & 7;
        v8f acc = {};
        for (int s = 0; s < 16; ++s) {
            int k = 4 * s + 2 * half;
            v2f a, bf;
            a.x = sc[(nT * 16 + l16) * NPAD + k];
            a.y = sc[(nT * 16 + l16) * NPAD + k + 1];
            bf.x = sV[(k)     * DD + dT * 16 + l16];
            bf.y = sV[(k + 1) * DD + dT * 16 + l16];
            acc = wmma_f32_step(a, bf, acc);
        }
#pragma unroll
        for (int j = 0; j < 8; ++j)
            S[(size_t)(b * NPAD + nT * 16 + j + 8 * half) * DD + dT * 16 + l16] = acc[j];
    }
}

// ---------------------------------------------------------------------------
// Kernel 3 (dominant GEMM, fused): m_scores = S @ region_emb^T /sqrt(D)
// + interp(E_d_match, hav) bias, softmax over n, weighted-sum -> out[b, r].
// One block (4 waves) per (batch, 16-region tile). S tile (32 KB) and
// region_emb tile (8 KB) are async-staged to LDS, overlapped with setup.
// No (B,N,R) intermediate is ever materialized.
// ---------------------------------------------------------------------------
__global__ __launch_bounds__(128) void k_match(
    const int* __restrict__ poi_idx,
    const float* __restrict__ lat, const float* __restrict__ lon,
    const float* __restrict__ cents, const float* __restrict__ Edm,
    const float* __restrict__ regE, const float* __restrict__ S,
    float* __restrict__ out)
{
    __shared__ float sS[NPAD * DD];                    // S[b] tile (32 KB)
    __shared__ float sR[16 * DD];                      // region tile (8 KB)
    __shared__ float ms[NPAD * 16];                    // 64 x 16 score column
    __shared__ float slat[NPAD], slon[NPAD], sval[NPAD];
    __shared__ float sE[64];
    __shared__ float clat[16], clon[16];

    const int b = blockIdx.y;
    const int rBase = blockIdx.x * 16;
    const int tid = threadIdx.x;

    // --- async stage S[b] (64x128) and region_emb[rBase:+16] into LDS ------
    {
        const float* ssrc = S + (size_t)b * NPAD * DD;
        for (int c = tid; c < (NPAD * DD) / 4; c += 128)       // 2048 x b128
            async_copy_b128(lds_addr_of(&sS[c * 4]), ssrc + c * 4);

        for (int c = tid; c < (16 * DD) / 4; c += 128) {       // 512 x b128
            int row = c >> 5;                                  // 32 chunks/row
            int cc  = (c & 31) * 4;
            int r = rBase + row;
            int rc = (r < RR) ? r : (RR - 1);                  // clamp; masked later
            async_copy_b128(lds_addr_of(&sR[row * DD + cc]),
                            regE + (size_t)rc * DD + cc);
        }
    }

    // setup (overlaps the async copies)
    if (tid < 64) {
        sE[tid] = Edm[tid];
        float vl = 0.f, la = 0.f, lo = 0.f;
        if (tid < NN) {
            int pi = poi_idx[b * NN + tid];
            vl = (pi >= 0) ? 1.0f : 0.0f;
            la = lat[b * NN + tid];
            lo = lon[b * NN + tid];
        }
        sval[tid] = vl; slat[tid] = la; slon[tid] = lo;
    }
    if (tid < 16) {
        int r = rBase + tid;
        int rc = (r < RR) ? r : (RR - 1);
        clat[tid] = cents[rc * 2 + 0];
        clon[tid] = cents[rc * 2 + 1];
    }

    wait_async0();
    __syncthreads();

    const int wave = tid >> 5, lane = tid & 31;
    const int half = lane >> 4, l16 = lane & 15;

    // --- one 16(n) x 16(r) WMMA tile per wave (nT = wave), K = 128 ---------
    {
        v8f acc = {};
        for (int s = 0; s < 32; ++s) {
            int k = 4 * s + 2 * half;
            v2f a, bf;
            a.x  = sS[(wave * 16 + l16) * DD + k];
            a.y  = sS[(wave * 16 + l16) * DD + k + 1];
            bf.x = sR[l16 * DD + k];                   // B[k][r] = regE[r][k]
            bf.y = sR[l16 * DD + k + 1];
            acc = wmma_f32_step(a, bf, acc);
        }
#pragma unroll
        for (int j = 0; j < 8; ++j)
            ms[(wave * 16 + j + 8 * half) * 16 + l16] = acc[j] * RSQRT_D;
    }
    __syncthreads();

    // --- haversine bias vs centroid + n-mask (TRANS units overlap XDL) -----
    for (int i = 0; i < 8; ++i) {
        int idx = tid + i * 128;
        int n = idx >> 4, rc = idx & 15;
        bool vn = (n < NN) && (sval[n] > 0.0f);
        if (vn) {
            float dd = hav_km(slat[n], slon[n], clat[rc], clon[rc]);
            float db = fminf(fmaxf(dd, 0.0f), DMAXC) / DMAXC * 63.0f;
            ms[idx] += interp64(sE, db);
        } else {
            ms[idx] = NEG_BIG;
        }
    }
    __syncthreads();

    // --- column softmax over n and weighted reduction -> out[b, r] ---------
    if (tid < 16) {
        int r = rBase + tid;
        if (r < RR) {
            float mx = NEG_BIG;
            for (int n = 0; n < NPAD; ++n) mx = fmaxf(mx, ms[n * 16 + tid]);
            float sum = 0.0f, num = 0.0f;
            for (int n = 0; n < NPAD; ++n) {
                float v = ms[n * 16 + tid];
                float e = __expf(v - mx);               // masked -> exp(-1e30) == 0
                sum += e;
                num += e * v;
            }
            out[(size_t)b * RR + r] = (sum > 0.0f && mx > -1e29f) ? (num / sum) : 0.0f;
        }
    }
}

// ---------------------------------------------------------------------------
extern "C" void kernel_launch(void* const* d_in, const int* in_sizes, int n_in,
                              void* d_out, int out_size, void* d_ws, size_t ws_size,
                              hipStream_t stream) {
    (void)in_sizes; (void)n_in; (void)out_size; (void)ws_size;

    const int*   poi_idx  = (const int*)  d_in[0];
    const int*   hour     = (const int*)  d_in[1];
    const float* lat      = (const float*)d_in[2];
    const float* lon      = (const float*)d_in[3];
    const float* t_min    = (const float*)d_in[4];
    const float* cents    = (const float*)d_in[5];
    const float* poi_emb  = (const float*)d_in[6];
    const float* time_emb = (const float*)d_in[7];
    const float* Et       = (const float*)d_in[8];
    const float* Ed       = (const float*)d_in[9];
    const float* Edm      = (const float*)d_in[10];
    const float* regE     = (const float*)d_in[11];
    const float* Wq       = (const float*)d_in[12];
    const float* Wk       = (const float*)d_in[13];
    const float* Wv       = (const float*)d_in[14];

    const size_t matElems = (size_t)BB * NPAD * DD;   // 8,388,608 floats each
    float* Q  = (float*)d_ws;
    float* Km = Q  + matElems;
    float* V  = Km + matElems;
    float* S  = V  + matElems;
    float* out = (float*)d_out;

    // 1) embed + QKV projection: 65536 padded rows / 16 per block
    k_embed_qkv<<<(BB * NPAD) / 16, 128, 0, stream>>>(
        poi_idx, hour, poi_emb, time_emb, Wq, Wk, Wv, Q, Km, V);

    // 2) per-batch attention (async V staging)
    k_attn<<<BB, 256, 0, stream>>>(
        poi_idx, lat, lon, t_min, Et, Ed, Q, Km, V, S);

    // 3) fused matcher GEMM + bias + softmax-over-n + reduction (async S/regE)
    k_match<<<dim3((RR + 15) / 16, BB), 128, 0, stream>>>(
        poi_idx, lat, lon, cents, Edm, regE, S, out);
}